// UPGO_43379169689828
// MI455X (gfx1250) — compile-verified
//
#include <hip/hip_runtime.h>
#include <cstdint>
#include <cstddef>

// Problem shape (fixed by the reference setup_inputs).
#define T_DIM 64
#define B_DIM 512
#define N_DIM 4096
#define ROWS  (T_DIM * B_DIM)   // 32768 rows of N_DIM floats

// ---------------------------------------------------------------------------
// Kernel 1: UPGO reverse scan -> advantages[t,b] = rhos * (returns - v_t)
// Serial in T (64 steps), parallel over B (512 columns). Tiny kernel.
// ---------------------------------------------------------------------------
__global__ __launch_bounds__(256)
void upgo_adv_kernel(const float* __restrict__ rewards,      // (T,B)
                     const float* __restrict__ bv,           // (T+1,B)
                     const float* __restrict__ rhos,         // (T,B)
                     float* __restrict__ adv) {              // (T,B) out
  int b = blockIdx.x * blockDim.x + threadIdx.x;
  if (b >= B_DIM) return;

  // g_last = r[T-1] + v_tp1[T-1]
  float g = rewards[(T_DIM - 1) * B_DIM + b] + bv[T_DIM * B_DIM + b];
  adv[(T_DIM - 1) * B_DIM + b] =
      rhos[(T_DIM - 1) * B_DIM + b] * (g - bv[(T_DIM - 1) * B_DIM + b]);

  for (int t = T_DIM - 2; t >= 0; --t) {
    float r   = rewards[t * B_DIM + b];
    float v1  = bv[(t + 1) * B_DIM + b];                      // v_tp1[t]
    // lam[t] = (r[t+1] + v_tp1[t+1] >= v_t[t+1]) ? 1 : 0
    float lam = (rewards[(t + 1) * B_DIM + b] + bv[(t + 2) * B_DIM + b] >=
                 bv[(t + 1) * B_DIM + b]) ? 1.0f : 0.0f;
    g = r + lam * g + (1.0f - lam) * v1;
    adv[t * B_DIM + b] = rhos[t * B_DIM + b] * (g - bv[t * B_DIM + b]);
  }
}

// ---------------------------------------------------------------------------
// Kernel 2: per-row online logsumexp + gather, one 256-thread block per row.
// Row = 16 KB staged into LDS with CDNA5 async global->LDS B128 copies
// (ASYNCcnt), non-temporal (row data is streamed once, never reused -> keep
// the 192MB L2 clean). Then single-pass online max/sum-exp from LDS, wave32
// shuffle reduce, cross-wave LDS merge.
// partial[row] = adv[row] * (x[a] - m - log s).
// ---------------------------------------------------------------------------
__global__ __launch_bounds__(256)
void row_lse_kernel(const float* __restrict__ x,             // (ROWS, N)
                    const long long* __restrict__ action,    // (ROWS) int64
                    const float* __restrict__ adv,           // (ROWS)
                    float* __restrict__ partial) {           // (ROWS) out
  __shared__ __align__(16) float tile[N_DIM];                // 16 KB row stage
  __shared__ float wm[8];
  __shared__ float wsum[8];

  const int row = blockIdx.x;
  const int tid = threadIdx.x;
  const char* rowp = (const char*)(x + (size_t)row * N_DIM);

  // LDS byte offset of the tile (low 32 bits of the flat shared address are
  // the workgroup-relative LDS offset on CDNA5 flat addressing).
  const uint32_t lds_base = (uint32_t)(uintptr_t)&tile[0];

  // Stage the whole row: thread tid copies 16 bytes at byte offset
  // 16*tid + 4096*k, k = 0..3. Wave-coalesced 512B per instruction.
  // Tracked by ASYNCcnt; no VGPR destination. NT: streamed-once data.
#pragma unroll
  for (int k = 0; k < 4; ++k) {
    uint32_t lds_off = lds_base + (uint32_t)(tid * 16 + k * 4096);
    const void* gptr = (const void*)(rowp + tid * 16 + k * 4096);
    asm volatile("global_load_async_to_lds_b128 %0, %1, off th:TH_LOAD_NT"
                 :: "v"(lds_off), "v"(gptr)
                 : "memory");
  }
  // Drain this wave's async copies, then make all waves' data visible
  // (needed because thread 0 later reads tile[action] staged by any wave).
  asm volatile("s_wait_asynccnt 0x0" ::: "memory");
  __syncthreads();

  // Online (max, sumexp) over this thread's own 16 staged values.
  // Readback layout == staging layout; ds_read_b128 is bank-conflict-free
  // (lanes 0..15 sweep all 64 banks, lanes 16..31 the second phase).
  float m = -__builtin_inff();
  float s = 0.0f;
#pragma unroll
  for (int k = 0; k < 4; ++k) {
    const float4 v4 = *reinterpret_cast<const float4*>(&tile[tid * 4 + k * 1024]);
    const float vals[4] = {v4.x, v4.y, v4.z, v4.w};
#pragma unroll
    for (int j = 0; j < 4; ++j) {
      float v  = vals[j];
      float nm = fmaxf(m, v);
      s = s * __expf(m - nm) + __expf(v - nm);
      m = nm;
    }
  }

  // Wave32 butterfly reduction of the (m, s) pair.
#pragma unroll
  for (int off = 16; off > 0; off >>= 1) {
    float om = __shfl_xor(m, off, 32);
    float os = __shfl_xor(s, off, 32);
    float nm = fmaxf(m, om);
    s = s * __expf(m - nm) + os * __expf(om - nm);
    m = nm;
  }

  const int wave = tid >> 5;
  if ((tid & 31) == 0) { wm[wave] = m; wsum[wave] = s; }
  __syncthreads();

  if (tid == 0) {
    float M = wm[0];
    float S = wsum[0];
#pragma unroll
    for (int w = 1; w < 8; ++w) {
      float om = wm[w], os = wsum[w];
      float nm = fmaxf(M, om);
      S = S * __expf(M - nm) + os * __expf(om - nm);
      M = nm;
    }
    int a = (int)action[row];                 // 0 <= a < N_DIM
    float xa = tile[a];                       // staged row value at the action
    float metric = xa - M - __logf(S);        // log_softmax(x)[a]
    partial[row] = adv[row] * metric;
  }
}

// ---------------------------------------------------------------------------
// Kernel 3a: deterministic stage-1 reduction. Block b sums the 256 contiguous
// partials [b*256, (b+1)*256) with a fixed-order LDS tree -> inter[b].
// 128 blocks run in parallel at full-device bandwidth.
// ---------------------------------------------------------------------------
__global__ __launch_bounds__(256)
void reduce_stage1_kernel(const float* __restrict__ partial,
                          float* __restrict__ inter) {
  __shared__ float sm[256];
  sm[threadIdx.x] = partial[blockIdx.x * 256 + threadIdx.x];
  __syncthreads();
  for (int off = 128; off > 0; off >>= 1) {
    if ((int)threadIdx.x < off) sm[threadIdx.x] += sm[threadIdx.x + off];
    __syncthreads();
  }
  if (threadIdx.x == 0) inter[blockIdx.x] = sm[0];
}

// ---------------------------------------------------------------------------
// Kernel 3b: deterministic stage-2 reduction of 128 intermediates -> scalar.
// out = -(mean over T*B).
// ---------------------------------------------------------------------------
__global__ __launch_bounds__(128)
void reduce_stage2_kernel(const float* __restrict__ inter,
                          float* __restrict__ out) {
  __shared__ float sm[128];
  sm[threadIdx.x] = inter[threadIdx.x];
  __syncthreads();
  for (int off = 64; off > 0; off >>= 1) {
    if ((int)threadIdx.x < off) sm[threadIdx.x] += sm[threadIdx.x + off];
    __syncthreads();
  }
  if (threadIdx.x == 0) out[0] = -sm[0] / (float)ROWS;
}

// ---------------------------------------------------------------------------
// Host-side launcher.
// Inputs (setup_inputs order):
//   d_in[0] target_output (T,B,N) f32
//   d_in[1] rhos          (T,B)   f32
//   d_in[2] action        (T,B)   i64
//   d_in[3] rewards       (T,B)   f32
//   d_in[4] bootstrap     (T+1,B) f32
// Output: 1 f32 scalar.
// Workspace layout (floats): [0,32768) advantages, [32768,65536) per-row
// partials, [65536,65664) stage-1 intermediates.  ~257 KB total.
// ---------------------------------------------------------------------------
extern "C" void kernel_launch(void* const* d_in, const int* in_sizes, int n_in,
                              void* d_out, int out_size, void* d_ws, size_t ws_size,
                              hipStream_t stream) {
  (void)in_sizes; (void)n_in; (void)out_size; (void)ws_size;

  const float*      target  = (const float*)d_in[0];
  const float*      rhos    = (const float*)d_in[1];
  const long long*  action  = (const long long*)d_in[2];
  const float*      rewards = (const float*)d_in[3];
  const float*      bv      = (const float*)d_in[4];

  float* adv     = (float*)d_ws;
  float* partial = adv + ROWS;
  float* inter   = partial + ROWS;
  float* out     = (float*)d_out;

  upgo_adv_kernel<<<dim3((B_DIM + 255) / 256), dim3(256), 0, stream>>>(rewards, bv, rhos, adv);
  row_lse_kernel<<<dim3(ROWS), dim3(256), 0, stream>>>(target, action, adv, partial);
  reduce_stage1_kernel<<<dim3(ROWS / 256), dim3(256), 0, stream>>>(partial, inter);
  reduce_stage2_kernel<<<dim3(1), dim3(128), 0, stream>>>(inter, out);
}